// GRUDecoderCellDecoder_52381421142278
// MI455X (gfx1250) — compile-verified
//
#include <hip/hip_runtime.h>
#include <hip/hip_bf16.h>
#include <math.h>

#define B_    32
#define S_    4096
#define D_    1024
#define VOUT_ 32003
#define H3_   3072
#define SCHUNK 32               // positions per attention block (128 KiB LDS tile)
#define NCHUNK (S_ / SCHUNK)    // 128
#define SUBT   (SCHUNK / 16)    // 2 TDM sub-tiles of 16 rows each

typedef float v2f __attribute__((ext_vector_type(2)));
typedef float v8f __attribute__((ext_vector_type(8)));
typedef unsigned int u32x4 __attribute__((ext_vector_type(4)));
typedef int i32x4 __attribute__((ext_vector_type(4)));
typedef int i32x8 __attribute__((ext_vector_type(8)));

// ---------------------------------------------------------------------------
// Embedding gather: embs[b, :] = emb[tokens[b], :]
// ---------------------------------------------------------------------------
__global__ void gather_emb(const int* __restrict__ tokens,
                           const float* __restrict__ emb,
                           float* __restrict__ embs) {
    int idx = blockIdx.x * blockDim.x + threadIdx.x;   // 32*1024
    int b = idx >> 10;
    int d = idx & (D_ - 1);
    embs[idx] = emb[(size_t)tokens[b] * D_ + d];
}

// ---------------------------------------------------------------------------
// Generic M=32 GEMM:  C[m,n] = sum_k A[m,k] * B[n,k] + bias[n]
// fp32 WMMA 16x16x4. Each wave: one 16-wide N tile, both 16-row M tiles.
// ---------------------------------------------------------------------------
__global__ __launch_bounds__(256)
void wmma_gemm_m32(const float* __restrict__ A, int lda,
                   const float* __restrict__ B, int ldb,
                   const float* __restrict__ bias,
                   float* __restrict__ C, int ldc,
                   int N, int K) {
    const int wave  = threadIdx.x >> 5;
    const int lane  = threadIdx.x & 31;
    const int ntile = blockIdx.x * 8 + wave;
    const int n0    = ntile * 16;
    if (n0 >= N) return;                 // wave-uniform exit: EXEC stays all-1s

    const int half = lane >> 4;          // 0: lanes 0-15, 1: lanes 16-31
    const int lid  = lane & 15;
    const int ncol = n0 + lid;
    const int nb   = (ncol < N) ? ncol : (N - 1);   // clamp B loads at N edge
    const int kh   = half * 2;

    const float* __restrict__ Arow0 = A + (size_t)lid * lda;
    const float* __restrict__ Arow1 = A + (size_t)(16 + lid) * lda;
    const float* __restrict__ Brow  = B + (size_t)nb * ldb;

    v8f c0 = {0.f, 0.f, 0.f, 0.f, 0.f, 0.f, 0.f, 0.f};
    v8f c1 = {0.f, 0.f, 0.f, 0.f, 0.f, 0.f, 0.f, 0.f};

    for (int k = 0; k < K; k += 4) {
        if ((k & 255) == 0)
            __builtin_prefetch(Brow + k + 256, 0, 1);   // global_prefetch_b8
        v2f a0 = *(const v2f*)(Arow0 + k + kh);
        v2f a1 = *(const v2f*)(Arow1 + k + kh);
        v2f bb = *(const v2f*)(Brow  + k + kh);
        c0 = __builtin_amdgcn_wmma_f32_16x16x4_f32(false, a0, false, bb,
                                                   (short)0, c0, false, false);
        c1 = __builtin_amdgcn_wmma_f32_16x16x4_f32(false, a1, false, bb,
                                                   (short)0, c1, false, false);
    }

    if (ncol < N) {
        const float bv = bias ? bias[ncol] : 0.f;
        float* Cc = C + ncol;
        #pragma unroll
        for (int i = 0; i < 8; ++i) {
            int m = half * 8 + i;
            Cc[(size_t)m        * ldc] = c0[i] + bv;
            Cc[(size_t)(16 + m) * ldc] = c1[i] + bv;
        }
    }
}

// ---------------------------------------------------------------------------
// GRU pointwise gates with h_prev == 0 (so gh == bhh exactly, per reference)
// ---------------------------------------------------------------------------
__global__ void gru_pointwise(const float* __restrict__ gi,
                              const float* __restrict__ bhh,
                              float* __restrict__ h, int hstride) {
    int idx = blockIdx.x * blockDim.x + threadIdx.x;   // 32*1024
    int b = idx >> 10;
    int d = idx & (D_ - 1);
    const float* g = gi + (size_t)b * H3_;
    float ir = g[d], iz = g[D_ + d], in = g[2 * D_ + d];
    float r = 1.f / (1.f + expf(-(ir + bhh[d])));
    float z = 1.f / (1.f + expf(-(iz + bhh[D_ + d])));
    float n = tanhf(in + r * bhh[2 * D_ + d]);
    h[(size_t)b * hstride + d] = (1.f - z) * n;
}

// ---------------------------------------------------------------------------
// TDM helper: issue one 2D tensor_load_to_lds of 16 rows x 1024 f32 (64 KiB)
// D# group0: count=1 | lds_addr | global_addr | type=2
// D# group1: data_size=4B, tensor_dim0=1024, tensor_dim1=16, tile 1024x16,
//            tensor_dim0_stride=1024 elements.
// ---------------------------------------------------------------------------
#if __has_builtin(__builtin_amdgcn_tensor_load_to_lds)
#define HAVE_TDM 1
__device__ __forceinline__
void tdm_load_16rows(unsigned lds_addr, unsigned long long gaddr) {
    u32x4 g0;
    g0.x = 1u;                                     // count=1, no gather
    g0.y = lds_addr;                               // LDS byte address
    g0.z = (unsigned)gaddr;                        // global_addr[31:0]
    g0.w = ((unsigned)(gaddr >> 32) & 0x01FFFFFFu) // global_addr[56:32]
         | (2u << 30);                             // type=2 ("image")
    i32x8 g1;
    g1[0] = 0x00020000;        // workgroup_mask=0, data_size=2 (4 bytes)
    g1[1] = (int)(D_ << 16);   // tensor_dim0[15:0]=1024 (bits 63:48)
    g1[2] = (int)(16 << 16);   // tensor_dim0 hi=0, tensor_dim1[15:0]=16
    g1[3] = (int)(D_ << 16);   // tensor_dim1 hi=0, tile_dim0=1024
    g1[4] = 16;                // tile_dim1=16, tile_dim2=0
    g1[5] = D_;                // tensor_dim0_stride[31:0]=1024
    g1[6] = 0;                 // stride hi / tensor_dim1_stride lo
    g1[7] = 0;
    i32x4 gz = {0, 0, 0, 0};
#if defined(__clang_major__) && __clang_major__ >= 23
    i32x8 gz8 = {0, 0, 0, 0, 0, 0, 0, 0};
    __builtin_amdgcn_tensor_load_to_lds(g0, g1, gz, gz, gz8, 0);
#else
    __builtin_amdgcn_tensor_load_to_lds(g0, g1, gz, gz, 0);
#endif
}
#else
#define HAVE_TDM 0
#endif

// ---------------------------------------------------------------------------
// Attention, pass 1: per (b, 32-position chunk) partial softmax stats.
// enc chunk is staged LDS-resident once via TDM (single HBM pass over enc);
// both the dot pass and the weighted-sum pass read LDS.
// ---------------------------------------------------------------------------
__global__ __launch_bounds__(256)
void attn_partial(const float* __restrict__ hcat,
                  const float* __restrict__ enc,
                  const int* __restrict__ encmask,
                  float* __restrict__ pm, float* __restrict__ pl,
                  float* __restrict__ pacc) {
    const int b = blockIdx.y;
    const int chunk = blockIdx.x;
    const int s0 = chunk * SCHUNK;
    const int wave = threadIdx.x >> 5;
    const int lane = threadIdx.x & 31;

    __shared__ float tile[SCHUNK * D_];   // 128 KiB enc chunk
    __shared__ float sh_h[D_];
    __shared__ float sh_w[SCHUNK];
    __shared__ float s_m, s_l;

    const float* __restrict__ encb = enc + ((size_t)b * S_ + s0) * D_;

#if HAVE_TDM
    if (wave == 0) {                      // wave-uniform: one wave issues DMA
        const unsigned lds_base =
            (unsigned)(unsigned long long)(size_t)(void*)tile;
        const unsigned long long gbase = (unsigned long long)(size_t)encb;
        tdm_load_16rows(lds_base,                    gbase);
        tdm_load_16rows(lds_base + 16u * D_ * 4u,    gbase + 16ull * D_ * 4ull);
    }
#else
    for (int i = threadIdx.x; i < SCHUNK * D_; i += 256)
        tile[i] = encb[i];
#endif

    for (int i = threadIdx.x; i < D_; i += 256)
        sh_h[i] = hcat[(size_t)b * 2048 + i];

    // dot(h_t, enc_row) for one 16-row sub-tile (2 rows per wave)
    auto dot_subtile = [&](int i) {
        for (int s2 = wave; s2 < 16; s2 += 8) {
            const int s = i * 16 + s2;
            const float* row = tile + (size_t)s * D_;
            float acc = 0.f;
            #pragma unroll 4
            for (int k = lane; k < D_; k += 32) acc += sh_h[k] * row[k];
            #pragma unroll
            for (int off = 16; off > 0; off >>= 1)
                acc += __shfl_xor(acc, off, 32);
            if (lane == 0)
                sh_w[s] = (encmask[(size_t)b * S_ + s0 + s] == 0) ? -INFINITY
                                                                  : acc;
        }
    };

    // explicitly peeled 2-stage pipeline: compute on sub-tile 0 overlaps the
    // in-flight DMA of sub-tile 1 (s_wait_tensorcnt takes a literal imm)
#if HAVE_TDM
    if (wave == 0) __builtin_amdgcn_s_wait_tensorcnt(1);
#endif
    __syncthreads();                      // sub-tile 0 landed; sh_h ready
    dot_subtile(0);
#if HAVE_TDM
    if (wave == 0) __builtin_amdgcn_s_wait_tensorcnt(0);
#endif
    __syncthreads();                      // sub-tile 1 landed
    dot_subtile(1);
    __syncthreads();

    // chunk max + exp + sum (SCHUNK == 32: one element per lane)
    if (threadIdx.x < 32) {
        float m = sh_w[threadIdx.x];
        #pragma unroll
        for (int off = 16; off > 0; off >>= 1)
            m = fmaxf(m, __shfl_xor(m, off, 32));
        float p = (m == -INFINITY) ? 0.f : expf(sh_w[threadIdx.x] - m);
        sh_w[threadIdx.x] = p;
        float l = p;
        #pragma unroll
        for (int off = 16; off > 0; off >>= 1) l += __shfl_xor(l, off, 32);
        if (threadIdx.x == 0) { s_m = m; s_l = l; }
    }
    __syncthreads();
    if (threadIdx.x == 0) {
        pm[b * NCHUNK + chunk] = s_m;
        pl[b * NCHUNK + chunk] = s_l;
    }

    // weighted accumulation straight from the LDS-resident tile
    float a0 = 0.f, a1 = 0.f, a2 = 0.f, a3 = 0.f;
    const int d = threadIdx.x;
    for (int s = 0; s < SCHUNK; ++s) {
        const float p = sh_w[s];
        const float* row = tile + (size_t)s * D_;
        a0 += p * row[d];
        a1 += p * row[d + 256];
        a2 += p * row[d + 512];
        a3 += p * row[d + 768];
    }
    float* acc = pacc + (size_t)(b * NCHUNK + chunk) * D_;
    acc[d] = a0; acc[d + 256] = a1; acc[d + 512] = a2; acc[d + 768] = a3;
}

// ---------------------------------------------------------------------------
// Attention, pass 2: log-sum-exp merge of chunk partials -> summary into
// hcat[b, 1024:2048].
// ---------------------------------------------------------------------------
__global__ __launch_bounds__(256)
void attn_combine(const float* __restrict__ pm, const float* __restrict__ pl,
                  const float* __restrict__ pacc, float* __restrict__ hcat) {
    const int b = blockIdx.x;
    __shared__ float sw[NCHUNK];
    __shared__ float sL;
    if (threadIdx.x == 0) {
        float M = -INFINITY;
        for (int i = 0; i < NCHUNK; ++i) M = fmaxf(M, pm[b * NCHUNK + i]);
        float L = 0.f;
        for (int i = 0; i < NCHUNK; ++i) {
            float mi = pm[b * NCHUNK + i];
            float w = (M == -INFINITY || mi == -INFINITY) ? 0.f : expf(mi - M);
            sw[i] = w;
            L += pl[b * NCHUNK + i] * w;
        }
        sL = L;
    }
    __syncthreads();
    const float invL = (sL > 0.f) ? 1.f / sL : 0.f;
    const int t = threadIdx.x;
    #pragma unroll
    for (int c = 0; c < 4; ++c) {
        int d = t + 256 * c;
        float acc = 0.f;
        for (int i = 0; i < NCHUNK; ++i)
            acc += pacc[(size_t)(b * NCHUNK + i) * D_ + d] * sw[i];
        hcat[(size_t)b * 2048 + D_ + d] = acc * invL;
    }
}

// ---------------------------------------------------------------------------
extern "C" void kernel_launch(void* const* d_in, const int* in_sizes, int n_in,
                              void* d_out, int out_size, void* d_ws, size_t ws_size,
                              hipStream_t stream) {
    const int*   tokens  = (const int*)  d_in[0];
    const float* enc     = (const float*)d_in[1];
    const int*   encmask = (const int*)  d_in[2];
    const float* emb     = (const float*)d_in[3];
    const float* Wih0    = (const float*)d_in[4];
    // d_in[5] = Whh0 : multiplied by h0==0 in the reference -> unused
    const float* bih0    = (const float*)d_in[6];
    const float* bhh0    = (const float*)d_in[7];
    const float* Wih1    = (const float*)d_in[8];
    // d_in[9] = Whh1 : unused (h0==0)
    const float* bih1    = (const float*)d_in[10];
    const float* bhh1    = (const float*)d_in[11];
    const float* preoutW = (const float*)d_in[12];
    const float* preoutb = (const float*)d_in[13];
    const float* outW    = (const float*)d_in[14];
    const float* outb    = (const float*)d_in[15];
    float* logits = (float*)d_out;                    // (32, 1, 32003)

    // workspace layout (floats)
    float* ws     = (float*)d_ws;
    float* embs   = ws;                     // 32*1024
    float* gi     = embs   + B_ * D_;       // 32*3072 (reused for both layers)
    float* hbuf   = gi     + B_ * H3_;      // 32*1024
    float* hcat   = hbuf   + B_ * D_;       // 32*2048  [h_t | summary]
    float* preout = hcat   + B_ * 2048;     // 32*1024
    float* pm     = preout + B_ * D_;       // 32*128
    float* pl     = pm     + B_ * NCHUNK;   // 32*128
    float* pacc   = pl     + B_ * NCHUNK;   // 32*128*1024 (16 MiB)

    // 1) embedding gather
    gather_emb<<<(B_ * D_) / 256, 256, 0, stream>>>(tokens, emb, embs);

    // 2) GRU layer 0: gi0 = embs @ Wih0[:, :1024].T + bih0   (inps = [embs, 0])
    wmma_gemm_m32<<<H3_ / 16 / 8, 256, 0, stream>>>(
        embs, D_, Wih0, 2 * D_, bih0, gi, H3_, H3_, D_);
    gru_pointwise<<<(B_ * D_) / 256, 256, 0, stream>>>(gi, bhh0, hbuf, D_);

    // 3) GRU layer 1: gi1 = h @ Wih1.T + bih1 ; h_t -> hcat[:, 0:1024]
    wmma_gemm_m32<<<H3_ / 16 / 8, 256, 0, stream>>>(
        hbuf, D_, Wih1, D_, bih1, gi, H3_, H3_, D_);
    gru_pointwise<<<(B_ * D_) / 256, 256, 0, stream>>>(gi, bhh1, hcat, 2048);

    // 4) attention: single HBM pass over enc via TDM-staged LDS tiles
    attn_partial<<<dim3(NCHUNK, B_), 256, 0, stream>>>(
        hcat, enc, encmask, pm, pl, pacc);
    attn_combine<<<B_, 256, 0, stream>>>(pm, pl, pacc, hcat);

    // 5) preout: out = hcat @ preoutW.T + preoutb   (M=32, N=1024, K=2048)
    wmma_gemm_m32<<<(D_ / 16) / 8, 256, 0, stream>>>(
        hcat, 2048, preoutW, 2048, preoutb, preout, D_, D_, 2 * D_);

    // 6) logits = preout @ outW.T + outb            (M=32, N=32003, K=1024)
    const int ntiles = (VOUT_ + 15) / 16;
    wmma_gemm_m32<<<(ntiles + 7) / 8, 256, 0, stream>>>(
        preout, D_, outW, D_, outb, logits, VOUT_, VOUT_, D_);
}